// GCNConv_11235634447053
// MI455X (gfx1250) — compile-verified
//
#include <hip/hip_runtime.h>

#define IN_CH 128
#define OUT_CH 128
#define W_STRIDE 132  // padded LDS row stride (floats) to dodge bank conflicts

typedef __attribute__((ext_vector_type(2))) float v2f;
typedef __attribute__((ext_vector_type(8))) float v8f;

// ---------------- zero init: agg accumulator (=d_out) and deg ----------------
__global__ void zero_kernel(float4* __restrict__ out4, float4* __restrict__ deg4,
                            int nOut4, int nDeg4) {
    int i = blockIdx.x * blockDim.x + threadIdx.x;
    float4 z = make_float4(0.f, 0.f, 0.f, 0.f);
    if (i < nOut4) out4[i] = z;
    if (i < nDeg4) deg4[i] = z;
}

// ---------------- degree count over source (row) indices ----------------
__global__ void deg_kernel(const int* __restrict__ row, float* __restrict__ deg,
                           int n_edges) {
    int e = blockIdx.x * blockDim.x + threadIdx.x;
    if (e < n_edges) unsafeAtomicAdd(&deg[row[e]], 1.0f);
}

// ---------------- deg^{-1/2} with zero-guard ----------------
__global__ void dis_kernel(const float* __restrict__ deg, float* __restrict__ dis,
                           int n) {
    int i = blockIdx.x * blockDim.x + threadIdx.x;
    if (i < n) {
        float d = deg[i];
        dis[i] = (d > 0.f) ? rsqrtf(d) : 0.f;
    }
}

// ---------------- edge scatter: agg[row] += norm * x[col] ----------------
// One wave32 per edge, each lane owns 4 contiguous channels (float4 gather,
// 4 hardware f32 atomics). Gathered x row = 512B fully coalesced per wave.
__global__ __launch_bounds__(256) void scatter_kernel(
    const float* __restrict__ x, const int* __restrict__ row,
    const int* __restrict__ col, const float* __restrict__ dis,
    float* __restrict__ agg, int n_edges) {
    int tid = blockIdx.x * blockDim.x + threadIdx.x;
    int e = tid >> 5;
    if (e >= n_edges) return;
    int lane = tid & 31;
    int r = row[e];
    int c = col[e];
    float norm = dis[r] * dis[c];
    const float4 xv = *(const float4*)(x + (size_t)c * IN_CH + lane * 4);
    float* dst = agg + (size_t)r * IN_CH + lane * 4;
    unsafeAtomicAdd(dst + 0, norm * xv.x);
    unsafeAtomicAdd(dst + 1, norm * xv.y);
    unsafeAtomicAdd(dst + 2, norm * xv.z);
    unsafeAtomicAdd(dst + 3, norm * xv.w);
}

// ---------------- GEMM: out = agg @ W.T + b, via V_WMMA_F32_16X16X4_F32 ------
// In-place safe: each wave reads only its own 16-node strip of `agg` (fully
// consumed into accumulators across the k-loop) before storing those rows.
// A tile 16x4 f32: lane L -> row m=L&15, VGPR v holds K = k + 2*(L>>4) + v.
// B tile 4x16 f32: lane L -> col n=L&15, VGPR v holds K = k + 2*(L>>4) + v
//                  (B[k'][n] = W[o][k'], o = t*16 + n; read from padded LDS).
// D tile 16x16 f32: VGPR r -> row r + 8*(L>>4), col = L&15.
__global__ __launch_bounds__(256) void gemm_wmma_kernel(
    const float* agg, const float* __restrict__ W, const float* __restrict__ bias,
    float* out, int n_strips, int n_nodes) {
    extern __shared__ float sW[];  // [OUT_CH][W_STRIDE]

    for (int i = threadIdx.x; i < OUT_CH * IN_CH; i += 256) {
        sW[(i >> 7) * W_STRIDE + (i & 127)] = W[i];
    }
    __syncthreads();

    const int lane = threadIdx.x & 31;
    const int wave = threadIdx.x >> 5;
    const int strip = blockIdx.x * 8 + wave;
    if (strip >= n_strips) return;

    const int m = lane & 15;      // A row / B-D column within tile
    const int half = lane >> 4;   // lane half selects K+2 (A/B) and M+8 (D)

    v8f acc[8] = {};

    int anode = strip * 16 + m;
    if (anode > n_nodes - 1) anode = n_nodes - 1;  // clamp (keeps EXEC full)
    const float* arow = agg + (size_t)anode * IN_CH + 2 * half;

    for (int k = 0; k < IN_CH; k += 4) {
        v2f a = *(const v2f*)(arow + k);
#pragma unroll
        for (int t = 0; t < 8; ++t) {
            const float* bp = &sW[(t * 16 + m) * W_STRIDE + k + 2 * half];
            v2f b = *(const v2f*)bp;
            acc[t] = __builtin_amdgcn_wmma_f32_16x16x4_f32(
                false, a, false, b, (short)0, acc[t], false, false);
        }
    }

#pragma unroll
    for (int t = 0; t < 8; ++t) {
        float bv = bias[t * 16 + m];
#pragma unroll
        for (int r = 0; r < 8; ++r) {
            int rowNode = strip * 16 + r + 8 * half;
            if (rowNode < n_nodes)
                out[(size_t)rowNode * OUT_CH + t * 16 + m] = acc[t][r] + bv;
        }
    }
}

extern "C" void kernel_launch(void* const* d_in, const int* in_sizes, int n_in,
                              void* d_out, int out_size, void* d_ws, size_t ws_size,
                              hipStream_t stream) {
    (void)n_in; (void)out_size; (void)ws_size;

    const float* x  = (const float*)d_in[0];   // [N, 128] f32
    const int*   ei = (const int*)d_in[1];     // [2, E] (JAX default int32)
    const float* W  = (const float*)d_in[2];   // [128, 128] f32
    const float* b  = (const float*)d_in[3];   // [128] f32
    float* out = (float*)d_out;                // [N, 128] f32, doubles as agg

    const int n_nodes = in_sizes[0] / IN_CH;   // 50000
    const int n_edges = in_sizes[1] / 2;       // 600000
    const int* row = ei;                       // edge_index[0]
    const int* col = ei + n_edges;             // edge_index[1]

    float* deg = (float*)d_ws;                 // [n_nodes] f32
    float* dis = deg + ((n_nodes + 63) & ~63); // [n_nodes] f32

    // 1) zero agg accumulator (d_out) + deg
    int nOut4 = (n_nodes * IN_CH) / 4;
    int nDeg4 = (n_nodes + 3) / 4;
    zero_kernel<<<(nOut4 + 255) / 256, 256, 0, stream>>>(
        (float4*)out, (float4*)deg, nOut4, nDeg4);

    // 2) degree histogram (f32 hardware atomics)
    deg_kernel<<<(n_edges + 255) / 256, 256, 0, stream>>>(row, deg, n_edges);

    // 3) deg^{-1/2}
    dis_kernel<<<(n_nodes + 255) / 256, 256, 0, stream>>>(deg, dis, n_nodes);

    // 4) normalized scatter-add of messages (wave per edge)
    long long sthreads = (long long)n_edges * 32;
    scatter_kernel<<<(int)((sthreads + 255) / 256), 256, 0, stream>>>(
        x, row, col, dis, out, n_edges);

    // 5) WMMA f32 GEMM + bias, in place on d_out
    int n_strips = (n_nodes + 15) / 16;        // 3125
    gemm_wmma_kernel<<<(n_strips + 7) / 8, 256, OUT_CH * W_STRIDE * sizeof(float),
                      stream>>>(out, W, b, out, n_strips, n_nodes);
}